// CMA_21998822490492
// MI455X (gfx1250) — compile-verified
//
#include <hip/hip_runtime.h>
#include <stdint.h>

// ---------------------------------------------------------------------------
// CMA butterfly equalizer, strictly-sequential scan (T=100000 steps).
// Latency-bound: single wave32, lane j <-> tap j (31 live lanes).
// Input streamed global->LDS with gfx1250 async DMA (ASYNCcnt), 3-slot ring.
// Cross-lane reduction: 5x __shfl_xor (ds/permlane) per quantity.
// WMMA intentionally unused: K=4 f32 WMMA cannot shorten a 124-long
// serially-updated dot product's critical path (see analysis).
// ---------------------------------------------------------------------------

#define L_TAPS   31
#define N_SAMP   200031
#define T_STEPS  100000
#define CHUNK    1024            // samples per chunk per plane (dwords)
#define NSLOT    3               // triple buffer: read c, c+1 while DMAing c+2
#define RING     (CHUNK * NSLOT) // 3072 elements per plane
#define NCHUNK   196             // ceil((2*(T-1)+30+1)/CHUNK): covers t<=200028

__device__ __forceinline__ uint32_t lds_off(const void* p) {
  // Generic pointers to LDS: hardware uses addr[31:0] as the LDS offset
  // (CDNA5 ISA 10.2 aperture table), so the low 32 bits are the DS address.
  return (uint32_t)(uintptr_t)p;
}

__device__ __forceinline__ float wave32_sum(float v) {
  v += __shfl_xor(v, 1);
  v += __shfl_xor(v, 2);
  v += __shfl_xor(v, 4);
  v += __shfl_xor(v, 8);
  v += __shfl_xor(v, 16);
  return v;                      // full sum replicated to all 32 lanes
}

__global__ __launch_bounds__(32) void cma_scan_kernel(
    const float* __restrict__ yre, const float* __restrict__ yim,
    const float* __restrict__ tre, const float* __restrict__ tim,
    float* __restrict__ out) {
  // 4 planes: 0=y0.re 1=y0.im 2=y1.re 3=y1.im ; z[t] = y[(t-16) mod N]
  __shared__ __align__(16) float zbuf[4 * RING];   // 48 KB

  const int lane = threadIdx.x;
  const float* plane_src[4] = { yre, yim, yre + N_SAMP, yim + N_SAMP };

  // ---- async DMA of one chunk (32 x b128 per chunk) into ring slot c%3 ----
  auto issue_chunk = [&](int c) {
    const int slot = c % NSLOT;
#pragma unroll
    for (int a = 0; a < 4; ++a) {
      const float* sp = plane_src[a];
#pragma unroll
      for (int r = 0; r < CHUNK / 128; ++r) {       // 8 groups of 32 lanes x 4 dwords
        int tb = c * CHUNK + (r * 32 + lane) * 4;   // first t of this lane's group
        // z[t] = y[(t-16) mod N]; 16 % 4 == 0, so no 4-dword group straddles
        // the wrap and each group stays contiguous in global memory.
        int g = (tb < 16) ? (N_SAMP - 16 + tb) : (tb - 16);
        if (g > N_SAMP - 4) g = N_SAMP - 4;         // tail clamp (bytes never read)
        const float* gp = sp + g;
        uint32_t lo = lds_off(&zbuf[a * RING + slot * CHUNK + (tb - c * CHUNK)]);
        asm volatile("global_load_async_to_lds_b128 %0, %1, off"
                     :: "v"(lo), "v"(gp) : "memory");
      }
    }
  };

  // ---- taps resident in VGPRs: lane j holds tap j of all 4 filters --------
  float t0r = 0.f, t0i = 0.f, t1r = 0.f, t1i = 0.f;
  float t2r = 0.f, t2i = 0.f, t3r = 0.f, t3i = 0.f;
  if (lane < L_TAPS) {
    t0r = tre[0 * L_TAPS + lane];  t0i = tim[0 * L_TAPS + lane];
    t1r = tre[1 * L_TAPS + lane];  t1i = tim[1 * L_TAPS + lane];
    t2r = tre[2 * L_TAPS + lane];  t2i = tim[2 * L_TAPS + lane];
    t3r = tre[3 * L_TAPS + lane];  t3i = tim[3 * L_TAPS + lane];
  }

  issue_chunk(0);
  issue_chunk(1);
  asm volatile("s_wait_asynccnt 0" ::: "memory");

  const float mask = (lane < L_TAPS) ? 1.0f : 0.0f;  // lane31 contributes 0
  int jj = (lane < L_TAPS) ? lane : (L_TAPS - 1);
  int toff = (L_TAPS - 1) - jj;                      // ring index of z[2i+30-j]

  // preload window for step 0
  float w0r = mask * zbuf[0 * RING + toff];
  float w0i = mask * zbuf[1 * RING + toff];
  float w1r = mask * zbuf[2 * RING + toff];
  float w1i = mask * zbuf[3 * RING + toff];

  float  twolr  = 2.0f * 0.001f;   // 2*lr, halves every 20000 steps
  int    segLeft = 20000;
  int    i = 0;
  float2* out0 = (float2*)out;             // pol0: (o0r,o0i) at [i]
  float2* out1 = (float2*)out + T_STEPS;   // pol1: (o1r,o1i) at [i]

  for (int c = 0; c < NCHUNK; ++c) {
    if (c + 2 < NCHUNK) issue_chunk(c + 2);          // DMA overlaps compute
    int iend = (c + 1) * (CHUNK / 2);
    if (iend > T_STEPS) iend = T_STEPS;

    for (; i < iend; ++i) {
      // prefetch next step's samples (address-only dependency: hides DS latency
      // under this step's reduction chain)
      int toffn = toff + 2;  if (toffn >= RING) toffn -= RING;
      float n0r = mask * zbuf[0 * RING + toffn];
      float n0i = mask * zbuf[1 * RING + toffn];
      float n1r = mask * zbuf[2 * RING + toffn];
      float n1i = mask * zbuf[3 * RING + toffn];

      // butterfly partial products (lane j == tap j, sample z[2i+30-j])
      float p0r = t0r * w0r - t0i * w0i + t1r * w1r - t1i * w1i;
      float p0i = t0r * w0i + t0i * w0r + t1r * w1i + t1i * w1r;
      float p1r = t2r * w1r - t2i * w1i + t3r * w0r - t3i * w0i;
      float p1i = t2r * w1i + t2i * w1r + t3r * w0i + t3i * w0r;

      float o0r = wave32_sum(p0r);
      float o0i = wave32_sum(p0i);
      float o1r = wave32_sum(p1r);
      float o1i = wave32_sum(p1i);

      float e0 = 1.0f - (o0r * o0r + o0i * o0i);
      float e1 = 1.0f - (o1r * o1r + o1i * o1i);

      if (lane == 0) {
        out0[i] = make_float2(o0r, o0i);
        out1[i] = make_float2(o1r, o1i);
      }

      float c0r = twolr * e0 * o0r, c0i = twolr * e0 * o0i;
      float c1r = twolr * e1 * o1r, c1i = twolr * e1 * o1i;

      // taps += coef * conj(w); lane31 has w==0 so its taps remain 0
      t0r += c0r * w0r + c0i * w0i;   t0i += c0i * w0r - c0r * w0i;
      t1r += c0r * w1r + c0i * w1i;   t1i += c0i * w1r - c0r * w1i;
      t2r += c1r * w1r + c1i * w1i;   t2i += c1i * w1r - c1r * w1i;
      t3r += c1r * w0r + c1i * w0i;   t3i += c1i * w0r - c1r * w0i;

      if (--segLeft == 0) { segLeft = 20000; twolr *= 0.5f; }

      toff = toffn;
      w0r = n0r; w0i = n0i; w1r = n1r; w1i = n1i;
    }
    // chunk c+2 now resident before we start reading it next iteration
    asm volatile("s_wait_asynccnt 0" ::: "memory");
  }
}

extern "C" void kernel_launch(void* const* d_in, const int* in_sizes, int n_in,
                              void* d_out, int out_size, void* d_ws, size_t ws_size,
                              hipStream_t stream) {
  (void)in_sizes; (void)n_in; (void)out_size; (void)d_ws; (void)ws_size;
  const float* yre = (const float*)d_in[0];   // (2, N) flat
  const float* yim = (const float*)d_in[1];   // (2, N) flat
  const float* tre = (const float*)d_in[2];   // (4, 31) flat
  const float* tim = (const float*)d_in[3];   // (4, 31) flat
  float* out = (float*)d_out;                 // (2, T, 2) flat
  cma_scan_kernel<<<1, 32, 0, stream>>>(yre, yim, tre, tim, out);
}